// GNN_33621003993498
// MI455X (gfx1250) — compile-verified
//
#include <hip/hip_runtime.h>

typedef __attribute__((ext_vector_type(2))) float v2f;
typedef __attribute__((ext_vector_type(4))) float v4f;
typedef __attribute__((ext_vector_type(8))) float v8f;

#define NVAR    50000
#define NCON    50000
#define DIM     128
#define INIT_IN 32
#define IN_DIM  288          // 2*DIM + INIT_IN
#define DEG     16
#define MTILES  3125         // 50000 / 16
#define LDS_STRIDE 292       // 288 + 4 pad: rows land on 16 distinct bank pairs

static __device__ __forceinline__ v8f wmma_f32(v2f a, v2f b, v8f c) {
    // V_WMMA_F32_16X16X4_F32: D(16x16 f32) = A(16x4 f32) * B(4x16 f32) + C
    return __builtin_amdgcn_wmma_f32_16x16x4_f32(
        /*neg_a=*/false, a, /*neg_b=*/false, b,
        /*c_mod=*/(short)0, c, /*reuse_a=*/false, /*reuse_b=*/false);
}

// ---------------------------------------------------------------------------
// lv = x[0:NV]   @ W_iv^T + b_iv   (side 0)
// lc = x[NV:NV+NC] @ W_ic^T + b_ic (side 1)
// grid: 2*MTILES blocks of 256 (8 waves, one 16-col slab each), K = 32
// ---------------------------------------------------------------------------
__global__ __launch_bounds__(256) void gnn_init(
    const float* __restrict__ x,
    const float* __restrict__ Wiv, const float* __restrict__ biv,
    const float* __restrict__ Wic, const float* __restrict__ bic,
    float* __restrict__ lv, float* __restrict__ lc)
{
    const int side = blockIdx.x / MTILES;
    const int tm   = blockIdx.x % MTILES;
    const int wave = threadIdx.x >> 5;
    const int lane = threadIdx.x & 31;
    const int half = lane >> 4;
    const int nl   = lane & 15;
    const int n0   = wave * 16;

    const float* __restrict__ A    = x + (size_t)(side ? (NVAR + tm * 16) : (tm * 16)) * INIT_IN;
    const float* __restrict__ W    = side ? Wic : Wiv;
    const float* __restrict__ bias = side ? bic : biv;
    float*       __restrict__ out  = side ? lc  : lv;

    const float* arow = A + (size_t)nl * INIT_IN + 2 * half;
    const float* brow = W + (size_t)(n0 + nl) * INIT_IN + 2 * half;

    v8f acc = {};
    #pragma unroll
    for (int k = 0; k < INIT_IN; k += 4) {
        v2f a = *(const v2f*)(arow + k);
        v2f b = *(const v2f*)(brow + k);
        acc = wmma_f32(a, b, acc);
    }

    const float bn = bias[n0 + nl];
    const size_t outbase = (size_t)tm * 16 * DIM + n0 + nl;
    #pragma unroll
    for (int v = 0; v < 8; ++v)
        out[outbase + (size_t)(v + 8 * half) * DIM] = acc[v] + bn;
}

// ---------------------------------------------------------------------------
// One message-passing layer, both sides in one launch (both read only "old"):
//  side 0: lc_out = [gather_sum(lv_in, cvi) | lc_in | con_feat] @ Wc^T + bc
//  side 1: lv_out = [gather_sum(lc_in, vci) | lv_in | var_feat] @ Wv^T + bv
// Block = 256 threads handles one 16-row tile: stage the 16x288 concat input
// in LDS cooperatively, then 8 waves each WMMA one 16-col slab (72 k-steps).
// ---------------------------------------------------------------------------
__global__ __launch_bounds__(256) void gnn_layer(
    const float* __restrict__ lv_in, const float* __restrict__ lc_in,
    const float* __restrict__ x,
    const long long* __restrict__ vci,   // var_constr_index (NV,16)
    const long long* __restrict__ cvi,   // constr_var_index (NC,16)
    const float* __restrict__ Wv, const float* __restrict__ bv,
    const float* __restrict__ Wc, const float* __restrict__ bc,
    float* __restrict__ lv_out, float* __restrict__ lc_out)
{
    __shared__ float Ain[16][LDS_STRIDE];

    const int side = blockIdx.x / MTILES;
    const int tm   = blockIdx.x % MTILES;

    const float* __restrict__ aggsrc = side ? lc_in : lv_in;
    const long long* __restrict__ idx = side ? vci  : cvi;
    const float* __restrict__ mid    = side ? lv_in : lc_in;
    const float* __restrict__ feat   = x + (size_t)(side ? 0 : NVAR) * INIT_IN;
    const float* __restrict__ W      = side ? Wv : Wc;
    const float* __restrict__ bias   = side ? bv : bc;
    float*       __restrict__ out    = side ? lv_out : lc_out;

    // ---- stage 1: build 16 x [agg(128) | old(128) | feat(32)] tile in LDS
    const int tid = threadIdx.x;
    const int r   = tid >> 4;      // 0..15 tile row
    const int tc  = tid & 15;      // 0..15 col chunk (8 floats each)
    const size_t gr = (size_t)tm * 16 + r;

    {   // neighbor aggregation, cols [tc*8, tc*8+8)
        v4f s0 = {}, s1 = {};
        const long long* irow = idx + gr * DEG;
        #pragma unroll 4
        for (int j = 0; j < DEG; ++j) {
            const float* src = aggsrc + (size_t)irow[j] * DIM + tc * 8;
            s0 += *(const v4f*)(src);
            s1 += *(const v4f*)(src + 4);
        }
        *(v4f*)&Ain[r][tc * 8]     = s0;
        *(v4f*)&Ain[r][tc * 8 + 4] = s1;
    }
    {   // old-activation segment, cols 128..255
        const float* m = mid + gr * DIM + tc * 8;
        *(v4f*)&Ain[r][128 + tc * 8]     = *(const v4f*)(m);
        *(v4f*)&Ain[r][128 + tc * 8 + 4] = *(const v4f*)(m + 4);
    }
    if (tc < 4) {  // raw-feature segment, cols 256..287
        const float* f = feat + gr * INIT_IN + tc * 8;
        *(v4f*)&Ain[r][256 + tc * 8]     = *(const v4f*)(f);
        *(v4f*)&Ain[r][256 + tc * 8 + 4] = *(const v4f*)(f + 4);
    }
    __syncthreads();

    // ---- stage 2: 16x16 output tile per wave via f32 WMMA, K = 288
    const int wave = tid >> 5;
    const int lane = tid & 31;
    const int half = lane >> 4;
    const int nl   = lane & 15;
    const int n0   = wave * 16;

    const float* brow = W + (size_t)(n0 + nl) * IN_DIM + 2 * half;

    v8f acc = {};
    #pragma unroll 8
    for (int k = 0; k < IN_DIM; k += 4) {
        v2f a = *(const v2f*)&Ain[nl][k + 2 * half];
        v2f b = *(const v2f*)(brow + k);
        acc = wmma_f32(a, b, acc);
    }

    const float bn = bias[n0 + nl];
    const size_t outbase = (size_t)tm * 16 * DIM + n0 + nl;
    #pragma unroll
    for (int v = 0; v < 8; ++v)
        out[outbase + (size_t)(v + 8 * half) * DIM] = acc[v] + bn;
}

// ---------------------------------------------------------------------------
// g[c] = sum_r lv[r][c]
// ---------------------------------------------------------------------------
__global__ void gnn_zero(float* __restrict__ g) { g[threadIdx.x] = 0.0f; }

__global__ __launch_bounds__(128) void gnn_reduce(const float* __restrict__ lv,
                                                  float* __restrict__ g)
{
    const int c = threadIdx.x;          // 0..127 = column
    float acc = 0.0f;
    for (int r = blockIdx.x; r < NVAR; r += gridDim.x)
        acc += lv[(size_t)r * DIM + c];
    atomicAdd(&g[c], acc);
}

// ---------------------------------------------------------------------------
// Q[i] = b_q + dot(Wq[0:128], g) + dot(Wq[128:256], lv[i])
// ---------------------------------------------------------------------------
__global__ __launch_bounds__(256) void gnn_final(
    const float* __restrict__ lv, const float* __restrict__ g,
    const float* __restrict__ Wq, const float* __restrict__ bq,
    float* __restrict__ Q)
{
    const int i = blockIdx.x * 256 + threadIdx.x;
    if (i >= NVAR) return;
    float s = bq[0];
    #pragma unroll 4
    for (int c = 0; c < DIM; ++c) s += Wq[c] * g[c];
    const float* row = lv + (size_t)i * DIM;
    #pragma unroll 4
    for (int c = 0; c < DIM; ++c) s += Wq[DIM + c] * row[c];
    Q[i] = s;
}

// ---------------------------------------------------------------------------
extern "C" void kernel_launch(void* const* d_in, const int* in_sizes, int n_in,
                              void* d_out, int out_size, void* d_ws, size_t ws_size,
                              hipStream_t stream)
{
    (void)in_sizes; (void)n_in; (void)out_size; (void)ws_size;

    const float*     x   = (const float*)d_in[0];
    const long long* vci = (const long long*)d_in[1];   // var_constr_index int64
    const long long* cvi = (const long long*)d_in[2];   // constr_var_index int64
    const float* Wiv = (const float*)d_in[3];
    const float* biv = (const float*)d_in[4];
    const float* Wic = (const float*)d_in[5];
    const float* bic = (const float*)d_in[6];
    const float* Wv  = (const float*)d_in[7];
    const float* bv  = (const float*)d_in[8];
    const float* Wc  = (const float*)d_in[9];
    const float* bc  = (const float*)d_in[10];
    const float* Wq  = (const float*)d_in[11];
    const float* bq  = (const float*)d_in[12];

    float* ws  = (float*)d_ws;
    const size_t sz = (size_t)NVAR * DIM;
    float* lv0 = ws;
    float* lv1 = ws + sz;
    float* lc0 = ws + 2 * sz;
    float* lc1 = ws + 3 * sz;
    float* g   = ws + 4 * sz;

    const dim3 blk(256);
    gnn_init <<<2 * MTILES, blk, 0, stream>>>(x, Wiv, biv, Wic, bic, lv0, lc0);
    gnn_layer<<<2 * MTILES, blk, 0, stream>>>(lv0, lc0, x, vci, cvi, Wv, bv, Wc, bc, lv1, lc1);
    gnn_layer<<<2 * MTILES, blk, 0, stream>>>(lv1, lc1, x, vci, cvi, Wv, bv, Wc, bc, lv0, lc0);
    gnn_layer<<<2 * MTILES, blk, 0, stream>>>(lv0, lc0, x, vci, cvi, Wv, bv, Wc, bc, lv1, lc1);
    gnn_zero <<<1, 128, 0, stream>>>(g);
    gnn_reduce<<<256, 128, 0, stream>>>(lv1, g);
    gnn_final<<<(NVAR + 255) / 256, blk, 0, stream>>>(lv1, g, Wq, bq, (float*)d_out);
}